// UniBioseqPooler_53128745451526
// MI455X (gfx1250) — compile-verified
//
#include <hip/hip_runtime.h>

#define EOS_IDX 2

constexpr int B_ = 64;
constexpr int S_ = 2048;
constexpr int D_ = 1024;
constexpr int SCHUNK = 128;          // s rows per pooling block
constexpr int NCHUNK = S_ / SCHUNK;  // 16
constexpr int DEPTH  = 16;           // async pipeline depth (rows in flight per wave)
constexpr int BATCH  = 4;            // rows consumed per main-loop iteration

// Issue one 16B-per-lane async global->LDS copy (512B per wave, 4KB per block row).
// Non-temporal: the 512MB feature stream is read once and exceeds the 192MB L2.
#define UBP_ASYNC_B128(laddr, ga)                                              \
  asm volatile("global_load_async_to_lds_b128 %0, %1, off th:TH_LOAD_NT"       \
               :: "v"(laddr), "v"(ga) : "memory")

// ---------------------------------------------------------------------------
// Kernel 1: per-row last EOS position (max s where ids[b,s] == EOS, else 0)
// ---------------------------------------------------------------------------
__global__ void ubp_eos_kernel(const int* __restrict__ ids, int* __restrict__ last_eos) {
  const int b = blockIdx.x;
  int m = 0;
  for (int s = threadIdx.x; s < S_; s += 256) {
    if (ids[b * S_ + s] == EOS_IDX) m = s;  // s increases -> per-thread max
  }
  __shared__ int red[256];
  red[threadIdx.x] = m;
  __syncthreads();
  #pragma unroll
  for (int off = 128; off > 0; off >>= 1) {
    if ((int)threadIdx.x < off) {
      int o = red[threadIdx.x + off];
      if (o > red[threadIdx.x]) red[threadIdx.x] = o;
    }
    __syncthreads();
  }
  if (threadIdx.x == 0) last_eos[b] = red[0];
}

// ---------------------------------------------------------------------------
// Kernel 2: masked sum over an s-slice, streamed through LDS with
// global_load_async_to_lds_b128 (ASYNCcnt-pipelined, 16 rows in flight).
// Each thread owns 4 contiguous channels; each LDS slot region is
// thread-private, so only per-wave counter waits are needed (no barriers).
// ---------------------------------------------------------------------------
template <bool USE_ATOMIC>
__global__ void ubp_pool_kernel(const float* __restrict__ feat,
                                const int* __restrict__ last_eos,
                                float* __restrict__ dst) {
  const int b     = blockIdx.x;
  const int chunk = blockIdx.y;
  const int t     = threadIdx.x;  // 0..255
  const int last  = last_eos[b];

  int s0 = chunk * SCHUNK;          if (s0 < 1)    s0 = 1;
  int s1 = chunk * SCHUNK + SCHUNK; if (s1 > last) s1 = last;
  const int n = s1 - s0;            // rows this block must sum (uniform per block)

  __shared__ __align__(16) float tile[DEPTH][D_];  // 64 KB ring buffer

  float ax = 0.f, ay = 0.f, az = 0.f, aw = 0.f;
  const float* base = feat + ((unsigned long long)b * S_) * D_ + 4 * t;

  if (n > 0) {
    // Prologue: fill the pipeline.
    const int pre = n < DEPTH ? n : DEPTH;
    for (int p = 0; p < pre; ++p) {
      unsigned laddr = (unsigned)(unsigned long long)&tile[p][4 * t];
      unsigned long long ga =
          (unsigned long long)(base + (unsigned long long)(s0 + p) * D_);
      UBP_ASYNC_B128(laddr, ga);
    }

    int i = 0;
    // Main loop: DEPTH transfers outstanding at loop top; loads complete in
    // order, so asynccnt <= DEPTH-BATCH proves rows i..i+BATCH-1 have landed.
    for (; i + DEPTH <= n; i += BATCH) {
      asm volatile("s_wait_asynccnt 12" ::: "memory");  // DEPTH - BATCH
      #pragma unroll
      for (int j = 0; j < BATCH; ++j) {
        const int slot = (i + j) & (DEPTH - 1);
        const float4 v = *(const float4*)&tile[slot][4 * t];
        ax += v.x; ay += v.y; az += v.z; aw += v.w;
      }
      // Drain LDS reads before the async engine may overwrite these slots
      // (async LDS writes are unordered w.r.t. DS ops).
      asm volatile("s_wait_dscnt 0" ::: "memory");
      #pragma unroll
      for (int j = 0; j < BATCH; ++j) {
        const int nxt = i + DEPTH + j;
        if (nxt < n) {
          const int slot = (i + j) & (DEPTH - 1);
          unsigned laddr = (unsigned)(unsigned long long)&tile[slot][4 * t];
          unsigned long long ga =
              (unsigned long long)(base + (unsigned long long)(s0 + nxt) * D_);
          UBP_ASYNC_B128(laddr, ga);
        }
      }
    }
    // Epilogue: everything issued; wait for all and consume the tail.
    asm volatile("s_wait_asynccnt 0" ::: "memory");
    for (; i < n; ++i) {
      const int slot = i & (DEPTH - 1);
      const float4 v = *(const float4*)&tile[slot][4 * t];
      ax += v.x; ay += v.y; az += v.z; aw += v.w;
    }
  }

  if (USE_ATOMIC) {
    float* o = dst + (unsigned long long)b * D_ + 4 * t;
    atomicAdd(o + 0, ax); atomicAdd(o + 1, ay);
    atomicAdd(o + 2, az); atomicAdd(o + 3, aw);
  } else {
    // partial[chunk][b][d] — written unconditionally (zeros when n<=0),
    // so no pre-zeroing of the workspace is needed.
    float* o = dst + (((unsigned long long)chunk * B_ + b) * D_) + 4 * t;
    o[0] = ax; o[1] = ay; o[2] = az; o[3] = aw;
  }
}

// ---------------------------------------------------------------------------
// Kernel 3a (deterministic path): reduce the 16 partials in fixed order and
// divide by count = max(last_eos-1, 0).
// ---------------------------------------------------------------------------
__global__ void ubp_final_kernel(const float* __restrict__ partial,
                                 const int* __restrict__ last_eos,
                                 float* __restrict__ out) {
  const int b = blockIdx.x;
  const int t = threadIdx.x;  // 0..255
  float ax = 0.f, ay = 0.f, az = 0.f, aw = 0.f;
  #pragma unroll
  for (int c = 0; c < NCHUNK; ++c) {
    const float4 v = *(const float4*)(partial +
        (((unsigned long long)c * B_ + b) * D_) + 4 * t);
    ax += v.x; ay += v.y; az += v.z; aw += v.w;
  }
  int cnt = last_eos[b] - 1; if (cnt < 0) cnt = 0;
  const float inv = 1.0f / (float)cnt;  // cnt==0 -> inf; 0*inf = NaN == 0/0
  float* o = out + (unsigned long long)b * D_ + 4 * t;
  o[0] = ax * inv; o[1] = ay * inv; o[2] = az * inv; o[3] = aw * inv;
}

// ---------------------------------------------------------------------------
// Kernel 3b (atomic fallback path helpers)
// ---------------------------------------------------------------------------
__global__ void ubp_zero_kernel(float* __restrict__ out) {
  out[(unsigned long long)blockIdx.x * 1024u + threadIdx.x] = 0.0f;
}

__global__ void ubp_scale_kernel(float* __restrict__ out,
                                 const int* __restrict__ last_eos) {
  const int b = blockIdx.x;
  int cnt = last_eos[b] - 1; if (cnt < 0) cnt = 0;
  out[(unsigned long long)b * D_ + threadIdx.x] /= (float)cnt;
}

// ---------------------------------------------------------------------------
extern "C" void kernel_launch(void* const* d_in, const int* in_sizes, int n_in,
                              void* d_out, int out_size, void* d_ws, size_t ws_size,
                              hipStream_t stream) {
  (void)in_sizes; (void)n_in; (void)out_size;
  const float* feat = (const float*)d_in[0];
  const int*   ids  = (const int*)d_in[1];
  float*       out  = (float*)d_out;

  int* last = (int*)d_ws;  // 64 ints at workspace start
  const size_t PARTIAL_OFF   = 1024;
  const size_t PARTIAL_BYTES = (size_t)NCHUNK * B_ * D_ * sizeof(float);  // 4 MB

  ubp_eos_kernel<<<dim3(B_), dim3(256), 0, stream>>>(ids, last);

  if (ws_size >= PARTIAL_OFF + PARTIAL_BYTES) {
    // Deterministic reduction via workspace partials (fixed summation order).
    float* partial = (float*)((char*)d_ws + PARTIAL_OFF);
    ubp_pool_kernel<false><<<dim3(B_, NCHUNK), dim3(256), 0, stream>>>(feat, last, partial);
    ubp_final_kernel<<<dim3(B_), dim3(256), 0, stream>>>(partial, last, out);
  } else {
    // Fallback: float atomics directly into d_out.
    ubp_zero_kernel<<<dim3(B_), dim3(1024), 0, stream>>>(out);
    ubp_pool_kernel<true><<<dim3(B_, NCHUNK), dim3(256), 0, stream>>>(feat, last, out);
    ubp_scale_kernel<<<dim3(B_), dim3(1024), 0, stream>>>(out, last);
  }
}